// ScaleAdaptiveMoEFFN_9474697855376
// MI455X (gfx1250) — compile-verified
//
#include <hip/hip_runtime.h>
#include <math.h>

// ---------------------------------------------------------------------------
// ScaleAdaptive MoE FFN for MI455X (gfx1250), fp32 WMMA path + TDM staging.
// B=4 S=2048 D=1024 E=8 K=2 H=4096 SE=64. Weight-bandwidth-bound (256 MB of
// w1/w2 per call, ~14us floor at 23.3 TB/s) -> keep fp32 (matches reference)
// and use V_WMMA_F32_16X16X4_F32 for all matmul work.
// ---------------------------------------------------------------------------

typedef float v2f __attribute__((ext_vector_type(2)));
typedef float v8f __attribute__((ext_vector_type(8)));
typedef unsigned int u32x4 __attribute__((ext_vector_type(4)));
typedef int i32x4 __attribute__((ext_vector_type(4)));
typedef int i32x8 __attribute__((ext_vector_type(8)));

constexpr int Bc = 4, Sc = 2048, Dc = 1024, Ec = 8, Hc_ = 4096, SEc = 64;
constexpr int NTOK = Bc * Sc;       // 8192 tokens
constexpr int MT = 32;              // token tile per workgroup
constexpr int HCHUNK = 128;         // H chunk held in LDS between the two GEMMs
constexpr int XS_STRIDE = Dc + 4;   // pad: bank = (4m + k) % 64 -> conflict-free
constexpr int HS_STRIDE = HCHUNK + 4;
constexpr int NTHREADS = 512;       // 16 wave32

static __device__ __forceinline__ v8f wmma4(v2f a, v2f b, v8f c) {
  // 8 args: (neg_a, A, neg_b, B, c_mod, C, reuse_a, reuse_b)
  return __builtin_amdgcn_wmma_f32_16x16x4_f32(false, a, false, b, (short)0, c,
                                               false, false);
}

static __device__ __forceinline__ float gelu_exact(float x) {
  return 0.5f * x * (1.0f + erff(x * 0.70710678118654752f));
}

// ---------------------------------------------------------------------------
// Router: logits = [x | scale_emb] @ router_w, softmax, top-2, renormalize,
// scatter into dense [NTOK, E] dispatch-weight table in workspace.
// One wave32 per token.
// ---------------------------------------------------------------------------
__global__ __launch_bounds__(256)
void moe_router_kernel(const float* __restrict__ x,
                       const float* __restrict__ scale_emb,
                       const float* __restrict__ router_w,
                       const int* __restrict__ scale_idx,
                       float* __restrict__ disp) {
  const int wave = threadIdx.x >> 5;
  const int lane = threadIdx.x & 31;
  const int t = blockIdx.x * 8 + wave;

  float p[Ec];
#pragma unroll
  for (int e = 0; e < Ec; ++e) p[e] = 0.0f;

  const float* xr = x + (size_t)t * Dc;
  for (int d = lane; d < Dc; d += 32) {
    const float xv = xr[d];
    const float* r = router_w + (size_t)d * Ec;
#pragma unroll
    for (int e = 0; e < Ec; ++e) p[e] += xv * r[e];
  }
#pragma unroll
  for (int off = 16; off > 0; off >>= 1) {
#pragma unroll
    for (int e = 0; e < Ec; ++e) p[e] += __shfl_xor(p[e], off, 32);
  }

  if (lane == 0) {
    const int si = scale_idx[0];
    const float* sev = scale_emb + (size_t)si * SEc;
    for (int j = 0; j < SEc; ++j) {
      const float s = sev[j];
      const float* r = router_w + (size_t)(Dc + j) * Ec;
#pragma unroll
      for (int e = 0; e < Ec; ++e) p[e] += s * r[e];
    }
    float mx = p[0];
#pragma unroll
    for (int e = 1; e < Ec; ++e) mx = fmaxf(mx, p[e]);
    float pr[Ec], sum = 0.0f;
#pragma unroll
    for (int e = 0; e < Ec; ++e) { pr[e] = __expf(p[e] - mx); sum += pr[e]; }
    const float inv = 1.0f / sum;
#pragma unroll
    for (int e = 0; e < Ec; ++e) pr[e] *= inv;
    int i0 = 0;
#pragma unroll
    for (int e = 1; e < Ec; ++e) if (pr[e] > pr[i0]) i0 = e;
    int i1 = (i0 == 0) ? 1 : 0;
#pragma unroll
    for (int e = 0; e < Ec; ++e)
      if (e != i0 && pr[e] > pr[i1]) i1 = e;
    const float wsum = pr[i0] + pr[i1];
    float* dt = disp + (size_t)t * Ec;
#pragma unroll
    for (int e = 0; e < Ec; ++e) dt[e] = 0.0f;
    dt[i0] = pr[i0] / wsum;
    dt[i1] = pr[i1] / wsum;
  }
}

// ---------------------------------------------------------------------------
// TDM: copy the 32x1024 fp32 x-tile into LDS with padded row stride using the
// Tensor Data Mover in iterate mode (tile = one 1024-elt row, 32 iterations,
// lds_addr_increment = XS_STRIDE keeps the bank-conflict-free padding).
// Issued by one wave; tracked with TENSORcnt.
// ---------------------------------------------------------------------------
#if __has_builtin(__builtin_amdgcn_tensor_load_to_lds)
#define USE_TDM 1
static __device__ __forceinline__ void tdm_load_x_tile(const float* gsrc,
                                                       unsigned lds_byte_off) {
  const unsigned long long ga = (unsigned long long)(uintptr_t)gsrc;
  u32x4 g0;
  g0[0] = 1u;                                    // count=1, user mode
  g0[1] = lds_byte_off;                          // lds_addr (bytes)
  g0[2] = (unsigned)ga;                          // global_addr[31:0]
  g0[3] = ((unsigned)(ga >> 32) & 0x01FFFFFFu) | (2u << 30);  // addr hi | type=2
  i32x8 g1;
  g1[0] = (2 << 16) | (1 << 19);                 // data_size=4B, iterate_enable
  g1[1] = (int)((Dc & 0xFFFF) << 16);            // tensor_dim0[15:0]
  g1[2] = (int)((Dc >> 16) | ((MT & 0xFFFF) << 16));  // dim0 hi | tensor_dim1 lo
  g1[3] = (int)((Dc & 0xFFFF) << 16);            // tensor_dim1 hi=0 | tile_dim0
  g1[4] = 1;                                     // tile_dim1=1, tile_dim2=0
  g1[5] = Dc;                                    // tensor_dim0_stride[31:0]
  g1[6] = 0;
  g1[7] = 0;
  i32x4 g2;
  g2[0] = 0;
  g2[1] = XS_STRIDE;                             // lds_addr_increment (elts)
  g2[2] = Dc;                                    // global_addr_increment (elts)
  g2[3] = (MT - 1) << 16;                        // iterate_count: 31 -> 32 rows
  i32x4 g3 = {0, 0, 0, 0};
#if __clang_major__ >= 23
  i32x8 gpad = {0, 0, 0, 0, 0, 0, 0, 0};
  __builtin_amdgcn_tensor_load_to_lds(g0, g1, g2, g3, gpad, 0);
#else
  __builtin_amdgcn_tensor_load_to_lds(g0, g1, g2, g3, 0);
#endif
}
#else
#define USE_TDM 0
#endif

// ---------------------------------------------------------------------------
// Fused MoE FFN. One workgroup = 32-token tile, 16 waves.
//   loop e: loop H-chunks of 128:
//     stage1: h[32,128] = disp[:,e] * gelu(x @ w1[e][:,chunk] + b1)  (WMMA)
//     stage2: acc[32,1024] += h @ w2[e][chunk,:]                     (WMMA)
// Dispatch weight folded into h => expert-weighted sum accumulates directly
// in persistent WMMA C registers; single store, no atomics.
// B fragments are register-batched: one load clause feeds 8-16 WMMAs.
// ---------------------------------------------------------------------------
extern __shared__ __align__(16) float smem[];

__global__ __launch_bounds__(NTHREADS)
void moe_ffn_kernel(const float* __restrict__ x,
                    const float* __restrict__ w1,
                    const float* __restrict__ b1,
                    const float* __restrict__ w2,
                    const float* __restrict__ b2,
                    const float* __restrict__ disp,
                    float* __restrict__ out) {
  float* xs = smem;                          // [MT][XS_STRIDE]
  float* hs = xs + MT * XS_STRIDE;           // [MT][HS_STRIDE]
  float* dsh = hs + MT * HS_STRIDE;          // [MT][E]
  __shared__ int eflag;

  const int tid = threadIdx.x;
  const int wave = tid >> 5;
  const int lane = tid & 31;
  const int lhalf = lane >> 4;   // 0: K {0,1} / M 0..7 ; 1: K {2,3} / M 8..15
  const int l16 = lane & 15;
  const int tok0 = blockIdx.x * MT;

  // ---- stage 0: x tile -> padded LDS ----
#if USE_TDM
  if (tid < 32) {
    if (lane == 0)
      tdm_load_x_tile(x + (size_t)tok0 * Dc, (unsigned)(uintptr_t)xs);
    __builtin_amdgcn_s_wait_tensorcnt(0);
  }
#else
  for (int q = tid; q < MT * (Dc / 4); q += NTHREADS) {
    const int m = q >> 8;
    const int d4 = q & 255;
    const float4 v = ((const float4*)(x + (size_t)(tok0 + m) * Dc))[d4];
    *((float4*)&xs[m * XS_STRIDE + d4 * 4]) = v;
  }
#endif
  if (tid < MT * Ec) dsh[tid] = disp[(size_t)tok0 * Ec + tid];
  __syncthreads();

  // persistent accumulators: wave owns output cols [wave*64, wave*64+64)
  const int c0 = wave * 64;
  const v8f vzero = {0.f, 0.f, 0.f, 0.f, 0.f, 0.f, 0.f, 0.f};
  v8f acc[8];                                // [mt2(2)][nt2(4)]
#pragma unroll
  for (int i = 0; i < 8; ++i) acc[i] = vzero;

  // stage-1 tile ownership: 16 waves cover the 2x8 grid of 16x16 h-tiles
  const int mt = wave >> 3;                  // 0..1
  const int nt = wave & 7;                   // 0..7
  const int arow = mt * 16 + l16;

  for (int e = 0; e < Ec; ++e) {
    if (tid == 0) {
      int any = 0;
      for (int m = 0; m < MT; ++m) any |= (dsh[m * Ec + e] != 0.0f);
      eflag = any;
    }
    __syncthreads();
    if (!eflag) continue;                    // uniform: no token routed here

    const float* w1e = w1 + (size_t)e * Dc * Hc_;
    const float* w2e = w2 + (size_t)e * Hc_ * Dc;
    const float* b1e = b1 + (size_t)e * Hc_;

    for (int hc = 0; hc < Hc_; hc += HCHUNK) {
      if (hc + HCHUNK < Hc_)
        __builtin_prefetch(w2e + (size_t)(hc + HCHUNK) * Dc + c0 + lane * 4, 0, 1);

      // ---- stage 1: h tile = gelu(x @ w1 + b1) * disp ----
      const int hcol = hc + nt * 16 + l16;
      v8f c1 = vzero;
      for (int k = 0; k < Dc; k += 32) {     // 8 WMMA k-steps per load batch
        v2f aa[8], bb[8];
#pragma unroll
        for (int u = 0; u < 8; ++u) {
          const int kk = k + u * 4 + lhalf * 2;
          aa[u].x = xs[arow * XS_STRIDE + kk];
          aa[u].y = xs[arow * XS_STRIDE + kk + 1];
          bb[u].x = w1e[(size_t)kk * Hc_ + hcol];
          bb[u].y = w1e[(size_t)(kk + 1) * Hc_ + hcol];
        }
#pragma unroll
        for (int u = 0; u < 8; ++u) c1 = wmma4(aa[u], bb[u], c1);
      }
      const float bias1 = b1e[hcol];
#pragma unroll
      for (int v = 0; v < 8; ++v) {
        const int mloc = mt * 16 + v + 8 * lhalf;   // C layout: M = v + 8*half
        const float val = c1[v] + bias1;
        hs[mloc * HS_STRIDE + nt * 16 + l16] =
            gelu_exact(val) * dsh[mloc * Ec + e];
      }
      __syncthreads();

      // ---- stage 2: acc += h @ w2[chunk, c0:c0+64] ----
      for (int k2 = 0; k2 < HCHUNK; k2 += 8) {  // 2 k-steps, 16 WMMA per batch
        v2f a0[2], a1[2], bb[2][4];
#pragma unroll
        for (int u = 0; u < 2; ++u) {
          const int kk2 = k2 + u * 4 + lhalf * 2;
          a0[u].x = hs[l16 * HS_STRIDE + kk2];
          a0[u].y = hs[l16 * HS_STRIDE + kk2 + 1];
          a1[u].x = hs[(16 + l16) * HS_STRIDE + kk2];
          a1[u].y = hs[(16 + l16) * HS_STRIDE + kk2 + 1];
          const float* w2row = w2e + (size_t)(hc + kk2) * Dc + c0 + l16;
#pragma unroll
          for (int n = 0; n < 4; ++n) {
            bb[u][n].x = w2row[n * 16];
            bb[u][n].y = w2row[Dc + n * 16];
          }
        }
#pragma unroll
        for (int u = 0; u < 2; ++u) {
#pragma unroll
          for (int n = 0; n < 4; ++n) {
            acc[n]     = wmma4(a0[u], bb[u][n], acc[n]);
            acc[4 + n] = wmma4(a1[u], bb[u][n], acc[4 + n]);
          }
        }
      }
      __syncthreads();                       // hs reused by next chunk
    }
  }

  // ---- epilogue: weighted b2, store ----
#pragma unroll
  for (int mt2 = 0; mt2 < 2; ++mt2) {
#pragma unroll
    for (int nt2 = 0; nt2 < 4; ++nt2) {
      const int col = c0 + nt2 * 16 + l16;
#pragma unroll
      for (int v = 0; v < 8; ++v) {
        const int m = mt2 * 16 + v + 8 * lhalf;
        float val = acc[mt2 * 4 + nt2][v];
#pragma unroll
        for (int e = 0; e < Ec; ++e)
          val += dsh[m * Ec + e] * b2[(size_t)e * Dc + col];
        out[(size_t)(tok0 + m) * Dc + col] = val;
      }
    }
  }
}

// ---------------------------------------------------------------------------
extern "C" void kernel_launch(void* const* d_in, const int* in_sizes, int n_in,
                              void* d_out, int out_size, void* d_ws,
                              size_t ws_size, hipStream_t stream) {
  const float* x         = (const float*)d_in[0];
  const float* scale_emb = (const float*)d_in[1];
  const float* router_w  = (const float*)d_in[2];
  const float* w1        = (const float*)d_in[3];
  const float* b1        = (const float*)d_in[4];
  const float* w2        = (const float*)d_in[5];
  const float* b2        = (const float*)d_in[6];
  const int*   scale_idx = (const int*)d_in[7];
  float* out = (float*)d_out;
  float* disp = (float*)d_ws;                // [NTOK, E] dispatch weights

  moe_router_kernel<<<NTOK / 8, 256, 0, stream>>>(x, scale_emb, router_w,
                                                  scale_idx, disp);

  const size_t shmem =
      (size_t)(MT * XS_STRIDE + MT * HS_STRIDE + MT * Ec) * sizeof(float);
  moe_ffn_kernel<<<NTOK / MT, NTHREADS, shmem, stream>>>(x, w1, b1, w2, b2,
                                                         disp, out);
}